// EEG_Net_8_Stack_7679401525407
// MI455X (gfx1250) — compile-verified
//
#include <hip/hip_runtime.h>
#include <math.h>

// ---------------- types / WMMA helpers ----------------
typedef __attribute__((ext_vector_type(2))) float v2f;
typedef __attribute__((ext_vector_type(8))) float v8f;

__device__ __forceinline__ v8f wmma4(v2f a, v2f b, v8f c) {
#if defined(__gfx1250__)
  // D = A(16x4 f32) * B(4x16 f32) + C(16x16 f32)
  return __builtin_amdgcn_wmma_f32_16x16x4_f32(false, a, false, b, (short)0, c,
                                               false, false);
#else
  return c;
#endif
}

// A-matrix fragment (16x4 slice of row-major M, leading dim ld).
// lanes 0-15: rows r0+l, K = kk+0 / kk+1 ; lanes 16-31: K = kk+2 / kk+3
__device__ __forceinline__ v2f fragA(const float* M, int ld, int r0, int kk,
                                     int lane) {
  int l = lane & 15, hi = (lane >> 4) & 1;
  const float* p = M + (size_t)(r0 + l) * ld + kk + hi * 2;
  v2f r; r.x = p[0]; r.y = p[1];
  return r;
}

// B-matrix fragment where B[kr][col] = src[(kk+kr)*ld + c0+col] (row-major B)
__device__ __forceinline__ v2f fragB_rows(const float* Bm, int ld, int kk,
                                          int c0, int lane) {
  int l = lane & 15, hi = (lane >> 4) & 1;
  const float* p = Bm + (size_t)(kk + hi * 2) * ld + c0 + l;
  v2f r; r.x = p[0]; r.y = p[ld];
  return r;
}

// async global->LDS copy of one dword (CDNA5 ASYNCcnt path)
__device__ __forceinline__ void async_ld_b32(unsigned lds_addr,
                                             const void* gaddr) {
#if defined(__gfx1250__)
  asm volatile("global_load_async_to_lds_b32 %0, %1, off" ::"v"(lds_addr),
               "v"((unsigned long long)(uintptr_t)gaddr)
               : "memory");
#endif
}
__device__ __forceinline__ void async_wait0() {
#if defined(__gfx1250__)
  asm volatile("s_wait_asynccnt 0x0" ::: "memory");
#endif
}

// ---------------- constants ----------------
#define WS 224
#define NW 512          // 64 batch * 8 windows
__device__ __constant__ int c_fwd[7]  = {0, 2, 3, 4, 8, 9, 10};
__device__ __constant__ int c_back[7] = {14, 15, 16, 18, 19, 20, 21};
__device__ __constant__ int c_lr[8]   = {1, 5, 6, 7, 11, 12, 13, 17};

// shared-memory float offsets inside the fused kernel
#define ORI_OFF 0            // 4 buffers of 224*16
#define SCR_OFF 14336        // sAw(64*224)+sBw(64*224); stage3: 8x per-wave scratch
#define QKV_OFF 43008        // q,k,v 3*224*16 ; also aliases padded inputs
#define SMEM_FLOATS 53760    // 215040 bytes

// ---------------- slice x -> A1/B1/L1 in [n][c][t] layout ----------------
__global__ void slice_kernel(const float* __restrict__ x, float* A1, float* B1,
                             float* L1) {
  int idx = blockIdx.x * 256 + threadIdx.x;
  const int total = NW * 22 * WS;
  if (idx >= total) return;
  int t = idx % WS;
  int rest = idx / WS;
  int j = rest % 22;
  int n = rest / 22;
  size_t src = ((size_t)n * WS + t) * 22;
  if (j < 7) {
    A1[((size_t)n * 7 + j) * WS + t] = x[src + c_fwd[j]];
  } else if (j < 14) {
    B1[((size_t)n * 7 + (j - 7)) * WS + t] = x[src + c_back[j - 7]];
  } else {
    L1[((size_t)n * 8 + (j - 14)) * WS + t] = x[src + c_lr[j - 14]];
  }
}

// ---------------- per-(window,channel) training BatchNorm, in place --------
// layout p[n = b*8 + w][C][WS]; stats over b (64) and t (224)
__global__ void bn_win_kernel(float* p, int C, const float* g, const float* bt,
                              float eps) {
  int w = blockIdx.x / C, c = blockIdx.x % C;
  __shared__ float s1[256], s2[256];
  float a = 0.f, q = 0.f;
  for (int i = threadIdx.x; i < 64 * WS; i += 256) {
    int b = i / WS, t = i % WS;
    float v = p[(((size_t)b * 8 + w) * C + c) * WS + t];
    a += v; q += v * v;
  }
  s1[threadIdx.x] = a; s2[threadIdx.x] = q;
  __syncthreads();
  for (int st = 128; st > 0; st >>= 1) {
    if (threadIdx.x < st) {
      s1[threadIdx.x] += s1[threadIdx.x + st];
      s2[threadIdx.x] += s2[threadIdx.x + st];
    }
    __syncthreads();
  }
  const float invN = 1.0f / (64.0f * WS);
  float mean = s1[0] * invN;
  float var = s2[0] * invN - mean * mean;
  float sc = g[c] * rsqrtf(var + eps);
  float sh = bt[c] - mean * sc;
  for (int i = threadIdx.x; i < 64 * WS; i += 256) {
    int b = i / WS, t = i % WS;
    size_t o = (((size_t)b * 8 + w) * C + c) * WS + t;
    p[o] = p[o] * sc + sh;
  }
}

// ---------------- per-channel training BatchNorm on [B][C][HW], in place ----
__global__ void bn_nchw_kernel(float* p, int B, int C, int HW, const float* g,
                               const float* bt, float eps) {
  int c = blockIdx.x;
  __shared__ float s1[256], s2[256];
  float a = 0.f, q = 0.f;
  int N = B * HW;
  for (int i = threadIdx.x; i < N; i += 256) {
    int b = i / HW, r = i % HW;
    float v = p[((size_t)b * C + c) * HW + r];
    a += v; q += v * v;
  }
  s1[threadIdx.x] = a; s2[threadIdx.x] = q;
  __syncthreads();
  for (int st = 128; st > 0; st >>= 1) {
    if (threadIdx.x < st) {
      s1[threadIdx.x] += s1[threadIdx.x + st];
      s2[threadIdx.x] += s2[threadIdx.x + st];
    }
    __syncthreads();
  }
  float mean = s1[0] / N;
  float var = s2[0] / N - mean * mean;
  float sc = g[c] * rsqrtf(var + eps);
  float sh = bt[c] - mean * sc;
  for (int i = threadIdx.x; i < N; i += 256) {
    int b = i / HW, r = i % HW;
    size_t o = ((size_t)b * C + c) * HW + r;
    p[o] = p[o] * sc + sh;
  }
}

// ---------------- conv1d (64 -> nch, k=3, pad 1) + relu, LDS->LDS ----------
__device__ void c1d_relu(const float* src /*[64][224]*/, const float* wt,
                         const float* bias, int nch, float* dst /*[224][16]*/) {
  for (int idx = threadIdx.x; idx < WS * 16; idx += 256) {
    int t = idx >> 4, c = idx & 15;
    float r = 0.f;
    if (c < nch) {
      float acc = bias[c];
      for (int i = 0; i < 64; ++i) {
        const float* sr = src + i * WS;
        const float* wr = wt + (c * 64 + i) * 3;
        if (t >= 1) acc += sr[t - 1] * wr[0];
        acc += sr[t] * wr[1];
        if (t < WS - 1) acc += sr[t + 1] * wr[2];
      }
      r = fmaxf(acc, 0.f);
    }
    dst[idx] = r;
  }
}

// ---------------- single-head attention for one window (fused) -------------
__device__ void attn_block(float* smem, const float* qsrc /*[224][16]*/,
                           const float* kvsrc /*[224][16]*/, int E,
                           const float* inw, const float* inb, const float* ow,
                           const float* ob, float* __restrict__ outp, int n,
                           int chbase, int cOut) {
  float* q = smem + QKV_OFF;
  float* k = q + WS * 16;
  float* v = k + WS * 16;
  int tid = threadIdx.x;
  __syncthreads();  // protect q/k/v region from previous use
  float scale = rsqrtf((float)E);
  for (int idx = tid; idx < WS * 16; idx += 256) {
    int t = idx >> 4, e = idx & 15;
    float qv = 0.f, kv2 = 0.f, vv = 0.f;
    if (e < E) {
      float aq = inb[e], ak = inb[E + e], av = inb[2 * E + e];
      for (int f = 0; f < E; ++f) {
        float xq = qsrc[t * 16 + f];
        float xk = kvsrc[t * 16 + f];
        aq += xq * inw[e * E + f];
        ak += xk * inw[(E + e) * E + f];
        av += xk * inw[(2 * E + e) * E + f];
      }
      qv = aq * scale; kv2 = ak; vv = av;
    }
    q[idx] = qv; k[idx] = kv2; v[idx] = vv;
  }
  __syncthreads();

  int wave = tid >> 5, lane = tid & 31;
  float* srow = smem + SCR_OFF + wave * (16 * WS);  // per-wave [16][224]

  for (int jb = wave; jb < 14; jb += 8) {
    int r0 = jb * 16;
    // ---- S = q @ k^T, one 16-row stripe; q fragments hoisted across tiles --
    v2f aq0 = fragA(q, 16, r0, 0, lane);
    v2f aq1 = fragA(q, 16, r0, 4, lane);
    v2f aq2 = fragA(q, 16, r0, 8, lane);
    v2f aq3 = fragA(q, 16, r0, 12, lane);
    for (int ct = 0; ct < 14; ++ct) {
      v8f acc = {0.f, 0.f, 0.f, 0.f, 0.f, 0.f, 0.f, 0.f};
      acc = wmma4(aq0, fragA(k, 16, ct * 16, 0, lane), acc);
      acc = wmma4(aq1, fragA(k, 16, ct * 16, 4, lane), acc);
      acc = wmma4(aq2, fragA(k, 16, ct * 16, 8, lane), acc);
      acc = wmma4(aq3, fragA(k, 16, ct * 16, 12, lane), acc);
      int l = lane & 15, hi = lane >> 4;
#pragma unroll
      for (int r = 0; r < 8; ++r)
        srow[(r + hi * 8) * WS + ct * 16 + l] = acc[r];
    }
    // ---- softmax over 224 keys: each row split across lane / lane^16 ------
    {
      int r = lane & 15;
      int c0 = (lane >> 4) * 112;
      float* row = srow + r * WS;
      float m = -1e30f;
      for (int j = 0; j < 112; ++j) m = fmaxf(m, row[c0 + j]);
      m = fmaxf(m, __shfl_xor(m, 16, 32));
      float s = 0.f;
      for (int j = 0; j < 112; ++j) {
        float e = __expf(row[c0 + j] - m);
        row[c0 + j] = e;
        s += e;
      }
      s += __shfl_xor(s, 16, 32);
      float inv = 1.0f / s;
      for (int j = 0; j < 112; ++j) row[c0 + j] *= inv;
    }
    // ---- O = P @ V  (16 x 16, K = 224) ----
    v8f o = {0.f, 0.f, 0.f, 0.f, 0.f, 0.f, 0.f, 0.f};
    for (int kk = 0; kk < WS; kk += 4) {
      v2f a = fragA(srow, WS, 0, kk, lane);
      v2f b = fragB_rows(v, 16, kk, 0, lane);
      o = wmma4(a, b, o);
    }
    {  // stash O tile in srow[0..15][0..15] (stride 16)
      int l = lane & 15, hi = lane >> 4;
#pragma unroll
      for (int r = 0; r < 8; ++r) srow[(r + hi * 8) * 16 + l] = o[r];
    }
    // ---- output projection + store [n][chbase+c][t] ----
    for (int idx = lane; idx < 16 * E; idx += 32) {
      int r = idx / E, c = idx % E;
      float accp = ob[c];
      for (int e = 0; e < E; ++e) accp += srow[r * 16 + e] * ow[c * E + e];
      outp[((size_t)n * cOut + chbase + c) * WS + r0 + r] = accp;
    }
  }
}

// ---------------- fused xanet: conv2d(->64) + 4x conv1d + 2x attention -----
__global__ void __launch_bounds__(256)
xanet_kernel(const float* __restrict__ A, const float* __restrict__ Bm, int cA,
             int cB, const float* cAw, const float* cAb, const float* cBw,
             const float* cBb, const float* dAw, const float* dAb,
             const float* dBw, const float* dBb, const float* aAinw,
             const float* aAinb, const float* aAow, const float* aAob,
             const float* aBinw, const float* aBinb, const float* aBow,
             const float* aBob, float* __restrict__ outp, int cOut) {
  extern __shared__ float smem[];
  const int n = blockIdx.x;
  const int tid = threadIdx.x;

  float* A_ori  = smem + ORI_OFF;
  float* A_resh = A_ori + WS * 16;
  float* B_ori  = A_resh + WS * 16;
  float* B_resh = B_ori + WS * 16;
  float* sAw = smem + SCR_OFF;            // [64][224]
  float* sBw = sAw + 64 * WS;             // [64][224]
  float* sAin = smem + QKV_OFF;           // [16][226], aliases q/k/v region
  float* sBin = sAin + 16 * 226;

  if (tid == 0) {  // pull shared weights into GL2 early
    __builtin_prefetch(cAw, 0, 0);
    __builtin_prefetch(cBw, 0, 0);
    __builtin_prefetch(dAw, 0, 0);
    __builtin_prefetch(dBw, 0, 0);
    __builtin_prefetch(aAinw, 0, 0);
    __builtin_prefetch(aBinw, 0, 0);
  }

  // ---- stage 0: zero pads, async-copy interiors global -> LDS ----
  for (int idx = tid; idx < 16 * 226; idx += 256) {
    int c = idx / 226, tp = idx % 226;
    bool interior = (tp >= 1 && tp <= WS);
    if (!(interior && c < cA)) sAin[idx] = 0.f;
    if (!(interior && c < cB)) sBin[idx] = 0.f;
  }
  {
    unsigned ldsA = (unsigned)(uintptr_t)sAin;
    unsigned ldsB = (unsigned)(uintptr_t)sBin;
    for (int idx = tid; idx < cA * WS; idx += 256) {
      int c = idx / WS, t = idx % WS;
      async_ld_b32(ldsA + (unsigned)(c * 226 + t + 1) * 4u,
                   A + ((size_t)n * cA + c) * WS + t);
    }
    for (int idx = tid; idx < cB * WS; idx += 256) {
      int c = idx / WS, t = idx % WS;
      async_ld_b32(ldsB + (unsigned)(c * 226 + t + 1) * 4u,
                   Bm + ((size_t)n * cB + c) * WS + t);
    }
    async_wait0();
  }
  __syncthreads();

  // ---- stage 1: conv2d(3x3, W=1) == conv1d k=3 (middle column), relu ----
  for (int buf = 0; buf < 2; ++buf) {
    const float* src = buf ? sBin : sAin;
    const float* wgt = buf ? cBw : cAw;
    const float* bia = buf ? cBb : cAb;
    int nch = buf ? cB : cA;
    float* dst = buf ? sBw : sAw;
    for (int idx = tid; idx < 64 * WS; idx += 256) {
      int o = idx / WS, t = idx % WS;
      float acc = bia[o];
      for (int c = 0; c < nch; ++c) {
        const float* sr = src + c * 226 + t;  // sr[dk] == in[t+dk-1] (padded)
        const float* wr = wgt + ((size_t)o * nch + c) * 9 + 1;  // [.,.,dk,1]
        acc += sr[0] * wr[0] + sr[1] * wr[3] + sr[2] * wr[6];
      }
      dst[idx] = fmaxf(acc, 0.f);
    }
  }
  __syncthreads();

  // ---- stage 2: four conv1d heads (64 -> cA/cB), relu, [t][e16] layout ----
  c1d_relu(sAw, dAw, dAb, cA, A_ori);
  c1d_relu(sBw, dBw, dBb, cB, B_ori);
  c1d_relu(sAw, dBw, dBb, cB, A_resh);
  c1d_relu(sBw, dAw, dAb, cA, B_resh);
  __syncthreads();

  // ---- stage 3: attention A then B (q/k/v + scratch regions reused) ----
  attn_block(smem, A_ori, B_resh, cA, aAinw, aAinb, aAow, aAob, outp, n, 0,
             cOut);
  attn_block(smem, B_ori, A_resh, cB, aBinw, aBinb, aBow, aBob, outp, n, cA,
             cOut);
}

// ---------------- tail kernels ----------------
__global__ void conv1_kernel(const float* __restrict__ X2, const float* w,
                             const float* bb, float* t1) {
  int idx = blockIdx.x * 256 + threadIdx.x;
  if (idx >= 64 * 16 * WS) return;
  int t = idx % WS, o = (idx / WS) % 16, b = idx / (WS * 16);
  float acc = bb[o];
  for (int w8 = 0; w8 < 8; ++w8)
    for (int c = 0; c < 22; ++c)
      acc += X2[(((size_t)(b * 8 + w8)) * 22 + c) * WS + t] *
             w[(o * 8 + w8) * 22 + c];
  t1[idx] = acc > 0.f ? acc : expm1f(acc);
}

__global__ void conv2_kernel(const float* __restrict__ t1, const float* w,
                             const float* bb, float* t2) {
  int idx = blockIdx.x * 256 + threadIdx.x;
  if (idx >= 64 * 4 * 16 * 226) return;
  int x = idx % 226;
  int h = (idx / 226) % 16;
  int o = (idx / (226 * 16)) % 4;
  int b = idx / (226 * 16 * 4);
  float acc = bb[o];
  for (int kh = 0; kh < 2; ++kh) {
    int hh = h + kh;
    if (hh >= 16) continue;  // bottom zero pad row
    const float* row = t1 + ((size_t)b * 16 + hh) * WS;
    for (int kw = 0; kw < 32; ++kw) {
      int t = x + kw - 16;
      if (t >= 0 && t < WS) acc += row[t] * w[(o * 2 + kh) * 32 + kw];
    }
  }
  t2[idx] = acc > 0.f ? acc : expm1f(acc);
}

__global__ void pool_kernel(const float* __restrict__ in, float* out, int BC,
                            int H, int W, int wh, int ww, int sh, int sw,
                            int OH, int OW) {
  int idx = blockIdx.x * 256 + threadIdx.x;
  int total = BC * OH * OW;
  if (idx >= total) return;
  int j = idx % OW;
  int i = (idx / OW) % OH;
  int bc = idx / (OW * OH);
  float m = -3.402823466e38f;
  for (int a = 0; a < wh; ++a)
    for (int b = 0; b < ww; ++b)
      m = fmaxf(m, in[(size_t)bc * H * W + (i * sh + a) * W + (j * sw + b)]);
  out[idx] = m;
}

__global__ void conv3_kernel(const float* __restrict__ p1, const float* w,
                             const float* bb, float* t3) {
  int idx = blockIdx.x * 256 + threadIdx.x;
  if (idx >= 64 * 4 * 4 * 57) return;
  int j = idx % 57;
  int i = (idx / 57) % 4;
  int o = (idx / (57 * 4)) % 4;
  int b = idx / (57 * 4 * 4);
  float acc = bb[o];
  for (int ci = 0; ci < 4; ++ci)
    for (int kh = 0; kh < 8; ++kh) {
      int h = i + kh - 4;  // pad (4,3)
      if (h < 0 || h >= 4) continue;
      for (int kw = 0; kw < 4; ++kw) {
        int x = j + kw - 2;  // pad (2,1)
        if (x < 0 || x >= 57) continue;
        acc += p1[(((size_t)b * 4 + ci) * 4 + h) * 57 + x] *
               w[((o * 4 + ci) * 8 + kh) * 4 + kw];
      }
    }
  t3[idx] = acc > 0.f ? acc : expm1f(acc);
}

// ---------------- host launch ----------------
extern "C" void kernel_launch(void* const* d_in, const int* in_sizes, int n_in,
                              void* d_out, int out_size, void* d_ws,
                              size_t ws_size, hipStream_t stream) {
  const float* x = (const float*)d_in[0];
  const float** in = (const float**)d_in;

  float* ws = (float*)d_ws;
  float* A1 = ws;                       // 512*7*224
  float* B1 = A1 + 512 * 7 * WS;        // 512*7*224
  float* L1 = B1 + 512 * 7 * WS;        // 512*8*224
  float* X1 = L1 + 512 * 8 * WS;        // 512*14*224
  float* X2 = X1 + 512 * 14 * WS;       // 512*22*224
  float* t1 = X2 + 512 * 22 * WS;       // 64*16*224
  float* t2 = t1 + 64 * 16 * WS;        // 64*4*16*226
  float* p1 = t2 + 64 * 4 * 16 * 226;   // 64*4*4*57
  float* t3 = p1 + 64 * 4 * 4 * 57;     // 64*4*4*57

  const size_t SMEM = SMEM_FLOATS * sizeof(float);

  // slice x into forward / backward / LR channel groups
  slice_kernel<<<(NW * 22 * WS) / 256, 256, 0, stream>>>(x, A1, B1, L1);

  // xa1: per-window BN (eps 1e-5) then fused xanet
  bn_win_kernel<<<8 * 7, 256, 0, stream>>>(A1, 7, in[1], in[2], 1e-5f);
  bn_win_kernel<<<8 * 7, 256, 0, stream>>>(B1, 7, in[3], in[4], 1e-5f);
  xanet_kernel<<<NW, 256, SMEM, stream>>>(
      A1, B1, 7, 7, in[5], in[6], in[7], in[8], in[9], in[10], in[11], in[12],
      in[13], in[14], in[15], in[16], in[17], in[18], in[19], in[20], X1, 14);

  // xa2
  bn_win_kernel<<<8 * 14, 256, 0, stream>>>(X1, 14, in[21], in[22], 1e-5f);
  bn_win_kernel<<<8 * 8, 256, 0, stream>>>(L1, 8, in[23], in[24], 1e-5f);
  xanet_kernel<<<NW, 256, SMEM, stream>>>(
      X1, L1, 14, 8, in[25], in[26], in[27], in[28], in[29], in[30], in[31],
      in[32], in[33], in[34], in[35], in[36], in[37], in[38], in[39], in[40],
      X2, 22);

  // tail
  conv1_kernel<<<(64 * 16 * WS) / 256, 256, 0, stream>>>(X2, in[41], in[42],
                                                         t1);
  bn_nchw_kernel<<<16, 256, 0, stream>>>(t1, 64, 16, WS, in[43], in[44], 0.f);
  conv2_kernel<<<(64 * 4 * 16 * 226) / 256, 256, 0, stream>>>(t1, in[45],
                                                              in[46], t2);
  bn_nchw_kernel<<<4, 256, 0, stream>>>(t2, 64, 4, 16 * 226, in[47], in[48],
                                        0.f);
  pool_kernel<<<(64 * 4 * 4 * 57 + 255) / 256, 256, 0, stream>>>(
      t2, p1, 64 * 4, 16, 226, 2, 2, 4, 4, 4, 57);
  conv3_kernel<<<(64 * 4 * 4 * 57 + 255) / 256, 256, 0, stream>>>(p1, in[49],
                                                                  in[50], t3);
  bn_nchw_kernel<<<4, 256, 0, stream>>>(t3, 64, 4, 4 * 57, in[51], in[52],
                                        0.f);
  pool_kernel<<<(64 * 4 * 2 * 14 + 255) / 256, 256, 0, stream>>>(
      t3, (float*)d_out, 64 * 4, 4, 57, 2, 4, 2, 4, 2, 14);
}